// K_Smooth_Matching_38216619000503
// MI455X (gfx1250) — compile-verified
//
#include <hip/hip_runtime.h>
#include <math.h>

typedef __attribute__((ext_vector_type(2))) float v2f;
typedef __attribute__((ext_vector_type(8))) float v8f;

#define NPTS 8192
#define NB   4
#define KSEL 17          // GROUP_SIZE + 1 (self included)
#define WAVES_PER_BLOCK 8

// One wave32 handles a tile of 16 queries against all 8192 candidates.
// Distance tiles come from V_WMMA_F32_16X16X4_F32:
//   A (M=candidates, K=4) row  = [cx, cy, cz, sq_c]
//   B (K=4, N=queries)  col    = [-2qx, -2qy, -2qz, 1]
//   D[m][n] = sq_c - 2*dot(c,q);  add sq_q per column afterwards.
// Per ISA layouts (CDNA5 ISA):
//   A 16x4 f32: lanes 0-15 -> {K0,K1} in 2 VGPRs, lanes 16-31 -> {K2,K3}
//   B 4x16 f32: mirrored (lanes 0-15 -> {K0,K1}, lanes 16-31 -> {K2,K3})
//   D 16x16 f32: VGPR v, lanes 0-15 -> (M=v, N=lane); lanes 16-31 -> (M=v+8, N=lane-16)
// So lane l owns query (l&15) and candidates M = v + 8*(l>>4) of each tile;
// lane pair (n, n+16) covers all 16 rows and is merged at the end via LDS.

__global__ __launch_bounds__(256) void knn_topk_kernel(
    const float* __restrict__ xyz,   // [NB][NPTS][3]
    float* __restrict__ outDist,     // [NB*NPTS][16]
    int*   __restrict__ outIdx)      // [NB*NPTS][16]
{
    __shared__ float ldsD[WAVES_PER_BLOCK * 32 * KSEL];
    __shared__ int   ldsI[WAVES_PER_BLOCK * 32 * KSEL];

    const int lane  = threadIdx.x & 31;
    const int warp  = threadIdx.x >> 5;
    const int wgid  = blockIdx.x * WAVES_PER_BLOCK + warp;  // 0 .. NB*(NPTS/16)-1
    const int bIdx  = wgid >> 9;                            // / (NPTS/16)
    const int qbase = (wgid & 511) * 16;
    const int half  = lane >> 4;                            // 0 or 1
    const int l16   = lane & 15;

    const float* batchPtr = xyz + (size_t)bIdx * NPTS * 3;

    // ---- query operand (constant across the candidate sweep) ----
    const float* q = batchPtr + (size_t)(qbase + l16) * 3;
    const float qx = q[0], qy = q[1], qz = q[2];
    const float sqq = qx * qx + qy * qy + qz * qz;
    v2f bop;
    bop.x = half ? (-2.0f * qz) : (-2.0f * qx);
    bop.y = half ? 1.0f         : (-2.0f * qy);

    // ---- per-lane sorted top-17 (ascending) over this lane's half-tiles ----
    float sd[KSEL];
    int   si[KSEL];
#pragma unroll
    for (int i = 0; i < KSEL; ++i) { sd[i] = 3.0e38f; si[i] = 0; }

    for (int tile = 0; tile < NPTS / 16; ++tile) {
        const int ci = tile * 16 + l16;
        const float* c = batchPtr + (size_t)ci * 3;
        __builtin_prefetch(batchPtr + (size_t)(((tile + 8) & 511) * 16 + l16) * 3, 0, 0);

        const float cx = c[0], cy = c[1], cz = c[2];
        const float csq = cx * cx + cy * cy + cz * cz;
        v2f aop;
        aop.x = half ? cz  : cx;
        aop.y = half ? csq : cy;

        v8f acc = {};
        acc = __builtin_amdgcn_wmma_f32_16x16x4_f32(
            /*neg_a=*/false, aop, /*neg_b=*/false, bop,
            /*c_mod=*/(short)0, acc, /*reuse_a=*/false, /*reuse_b=*/false);

#pragma unroll
        for (int v = 0; v < 8; ++v) {
            float d2 = fmaxf(acc[v] + sqq, 0.0f);      // reference clamps before top_k
            int cand = tile * 16 + v + half * 8;
            if (d2 < sd[KSEL - 1]) {                   // cheap guard; rare full insert
                float dv = d2; int di = cand;
#pragma unroll
                for (int i = 0; i < KSEL; ++i) {
                    bool  lt = dv < sd[i];
                    float ns = lt ? dv : sd[i];
                    int   ni = lt ? di : si[i];
                    float od = sd[i]; int oi = si[i];
                    dv = lt ? od : dv;
                    di = lt ? oi : di;
                    sd[i] = ns; si[i] = ni;
                }
            }
        }
    }

    // ---- stash per-lane lists, then merge lane pairs (n, n+16) ----
    const int base = (warp * 32 + lane) * KSEL;
#pragma unroll
    for (int i = 0; i < KSEL; ++i) { ldsD[base + i] = sd[i]; ldsI[base + i] = si[i]; }
    __syncthreads();

    if (lane < 16) {
        const int aoff = (warp * 32 + lane) * KSEL;
        const int boff = (warp * 32 + lane + 16) * KSEL;
        const int qrow = bIdx * NPTS + qbase + lane;
        float* dout = outDist + (size_t)qrow * 16;
        int*   iout = outIdx  + (size_t)qrow * 16;

        int ia = 0, ib = 0;
        for (int r = 0; r < KSEL; ++r) {
            float da = ldsD[aoff + ia];
            float db = ldsD[boff + ib];
            bool takeA = (da <= db);
            float d = takeA ? da : db;
            int  ix = takeA ? ldsI[aoff + ia] : ldsI[boff + ib];
            ia += takeA ? 1 : 0;
            ib += takeA ? 0 : 1;
            if (r > 0) {                      // rank 0 is the self-match, dropped
                dout[r - 1] = sqrtf(fmaxf(d, 0.0f));
                iout[r - 1] = ix + bIdx * NPTS;
            }
        }
    }
}

extern "C" void kernel_launch(void* const* d_in, const int* in_sizes, int n_in,
                              void* d_out, int out_size, void* d_ws, size_t ws_size,
                              hipStream_t stream) {
    (void)in_sizes; (void)n_in; (void)d_ws; (void)ws_size; (void)out_size;
    const float* xyz = (const float*)d_in[0];
    float* outDist = (float*)d_out;                                  // dist first
    int*   outIdx  = (int*)d_out + (size_t)NB * NPTS * 16;           // then indices
    const int totalWaves = NB * (NPTS / 16);                         // 2048
    const int blocks = totalWaves / WAVES_PER_BLOCK;                 // 256
    knn_topk_kernel<<<blocks, 256, 0, stream>>>(xyz, outDist, outIdx);
}